// SSLLoss1_8804682956811
// MI455X (gfx1250) — compile-verified
//
#include <hip/hip_runtime.h>
#include <math.h>

typedef __attribute__((ext_vector_type(2))) float v2f;
typedef __attribute__((ext_vector_type(8))) float v8f;

#define DIM   64
#define KTOP  30
#define PANEL 2048
#define PSTR  2056          // +8 floats to break LDS bank alignment
#define NPANEL 3            // 3*2048 = 6144 >= 6000
#define CANDN 96            // 3*30 = 90 candidate slots (rounded up)
#define NEG_INF (-__builtin_inff())

// accumulator layout (doubles): [0..3]=E_self, [4..7]=T_topk, [8..9]=E12,
// [10..11]=M12, [12..13]=M21, [14..17]=SelfTerm
#define ACC_E   0
#define ACC_T   4
#define ACC_E12 8
#define ACC_M12 10
#define ACC_M21 12
#define ACC_ST  14

__device__ __forceinline__ double waveSumD(double v) {
  #pragma unroll
  for (int off = 16; off > 0; off >>= 1) v += __shfl_xor(v, off, 32);
  return v;
}
__device__ __forceinline__ float waveSumF(float v) {
  #pragma unroll
  for (int off = 16; off > 0; off >>= 1) v += __shfl_xor(v, off, 32);
  return v;
}

// ---------------- normalize: out[m] = row-L2-normalized input m ----------------
__global__ void k_normalize(const float* __restrict__ i0, const float* __restrict__ i1,
                            const float* __restrict__ i2, const float* __restrict__ i3,
                            float* __restrict__ out, int N) {
  const float* src = (blockIdx.y == 0) ? i0 : (blockIdx.y == 1) ? i1
                   : (blockIdx.y == 2) ? i2 : i3;
  float* dst = out + (size_t)blockIdx.y * N * DIM;
  int wid = threadIdx.x >> 5, lane = threadIdx.x & 31;
  int row = blockIdx.x * 8 + wid;
  if (row >= N) return;
  size_t base = (size_t)row * DIM + lane * 2;
  float x0 = src[base], x1 = src[base + 1];
  float ss = waveSumF(x0 * x0 + x1 * x1);
  float dnm = fmaxf(sqrtf(ss), 1e-12f);
  dst[base]     = x0 / dnm;
  dst[base + 1] = x1 / dnm;
}

__device__ __forceinline__ void load_afrag(const float* sA, int lane, v2f* afrag) {
  int ar = lane & 15;
  int koff = (lane < 16) ? 0 : 2;
  const float* p = sA + ar * DIM + koff;
  #pragma unroll
  for (int k16 = 0; k16 < 16; ++k16)
    afrag[k16] = *(const v2f*)(p + k16 * 4);
}

// compute one 16x16 tile of A(rows cached in afrag) x Brows^T via 16 chained WMMAs.
// Loads are unconditional: caller passes a clamped (always-valid) row pointer and
// masks out-of-range columns when consuming the accumulator.
__device__ __forceinline__ v8f tile16x16(const v2f* afrag, const float* Brow, int koff) {
  v2f bfrag[16];
  const float* p = Brow + koff;
  #pragma unroll
  for (int k16 = 0; k16 < 16; ++k16)
    bfrag[k16] = *(const v2f*)(p + k16 * 4);
  v8f c = {0.f, 0.f, 0.f, 0.f, 0.f, 0.f, 0.f, 0.f};
  #pragma unroll
  for (int k16 = 0; k16 < 16; ++k16)
    c = __builtin_amdgcn_wmma_f32_16x16x4_f32(false, afrag[k16], false, bfrag[k16],
                                              (short)0, c, false, false);
  return c;
}

// ---------------- self-sim: WMMA S=XX^T tiles -> LDS, exp-sum + per-row top-30 ----------------
__global__ void __launch_bounds__(512) k_selfsim(const float* __restrict__ norm, int N,
                                                 int* __restrict__ topkIdx,
                                                 double* __restrict__ acc) {
  extern __shared__ char smem[];
  float*  sims  = (float*)smem;                       // 16*PSTR floats
  float*  sA    = sims + 16 * PSTR;                   // 16*64
  float*  candV = sA + 16 * DIM;                      // 16*CANDN
  int*    candI = (int*)(candV + 16 * CANDN);         // 16*CANDN
  double* redD  = (double*)(candI + 16 * CANDN);      // 16

  int m = blockIdx.y;
  const float* X = norm + (size_t)m * N * DIM;
  int rowBase = blockIdx.x * 16;
  int tid = threadIdx.x, wid = tid >> 5, lane = tid & 31;

  for (int idx = tid; idx < 16 * DIM; idx += 512)
    sA[idx] = X[(size_t)(rowBase + (idx >> 6)) * DIM + (idx & 63)];
  __syncthreads();

  v2f afrag[16];
  load_afrag(sA, lane, afrag);
  int koff = (lane < 16) ? 0 : 2;

  double myE = 0.0;

  for (int p = 0; p < NPANEL; ++p) {
    int pb = p * PANEL;
    // --- compute 16 x PANEL sim block with f32 WMMA ---
    for (int t = wid; t < PANEL / 16; t += 16) {
      int cb = pb + t * 16;
      int jcol = cb + (lane & 15);
      bool jvalid = jcol < N;
      const float* Brow = X + (size_t)(jvalid ? jcol : 0) * DIM;  // clamped, loads safe
      v8f c = tile16x16(afrag, Brow, koff);
      int colL = t * 16 + (lane & 15);
      int rb = (lane < 16) ? 0 : 8;
      #pragma unroll
      for (int r = 0; r < 8; ++r)
        sims[(rb + r) * PSTR + colL] = jvalid ? c[r] : NEG_INF;
    }
    __syncthreads();

    // --- global exp-sum over this panel (fp64 accumulation) ---
    for (int r = 0; r < 16; ++r)
      for (int col = tid; col < PANEL; col += 512)
        if (pb + col < N) myE += (double)expf(sims[r * PSTR + col]);
    __syncthreads();

    // --- per-row top-30 within panel (wave wid owns row wid), destructive argmax ---
    {
      int row = wid;
      volatile float* srow = sims + row * PSTR;
      for (int sel = 0; sel < KTOP; ++sel) {
        float bv = NEG_INF; int bi = 0x7fffffff;
        for (int col = lane; col < PANEL; col += 32) {
          float v = srow[col];
          if (v > bv) { bv = v; bi = col; }
        }
        #pragma unroll
        for (int off = 16; off > 0; off >>= 1) {
          float ov = __shfl_xor(bv, off, 32);
          int   oi = __shfl_xor(bi, off, 32);
          if (ov > bv || (ov == bv && oi < bi)) { bv = ov; bi = oi; }
        }
        if (lane == 0) {
          candV[row * CANDN + p * KTOP + sel] = bv;
          candI[row * CANDN + p * KTOP + sel] = pb + bi;
          srow[bi] = NEG_INF;
        }
      }
    }
    __syncthreads();
  }

  // --- merge 90 candidates -> final top-30; emit indices + sum(exp(topvals)) ---
  {
    int row = wid;
    volatile float* cv = candV + row * CANDN;
    int* ci = candI + row * CANDN;
    double tsum = 0.0;
    int grow = rowBase + row;
    for (int sel = 0; sel < KTOP; ++sel) {
      float bv = NEG_INF; int bi = 0x7fffffff;
      for (int c2 = lane; c2 < NPANEL * KTOP; c2 += 32) {
        float v = cv[c2];
        if (v > bv) { bv = v; bi = c2; }
      }
      #pragma unroll
      for (int off = 16; off > 0; off >>= 1) {
        float ov = __shfl_xor(bv, off, 32);
        int   oi = __shfl_xor(bi, off, 32);
        if (ov > bv || (ov == bv && oi < bi)) { bv = ov; bi = oi; }
      }
      if (lane == 0) {
        topkIdx[((size_t)m * N + grow) * KTOP + sel] = ci[bi];
        tsum += (double)expf(bv);
        cv[bi] = NEG_INF;
      }
    }
    if (lane == 0) atomicAdd(&acc[ACC_T + m], tsum);
  }

  myE = waveSumD(myE);
  if (lane == 0) redD[wid] = myE;
  __syncthreads();
  if (tid == 0) {
    double s = 0.0;
    for (int w = 0; w < 16; ++w) s += redD[w];
    atomicAdd(&acc[ACC_E + m], s);
  }
}

// ---------------- cross: E12 = sum(exp(A1 A2^T)), straight from WMMA accumulators ----------------
__global__ void __launch_bounds__(512) k_cross(const float* __restrict__ norm, int N,
                                               double* __restrict__ acc) {
  __shared__ float sA[16 * DIM];
  __shared__ double redD[16];
  int d = blockIdx.y;
  const float* A1 = norm + (size_t)(2 * d)     * N * DIM;
  const float* A2 = norm + (size_t)(2 * d + 1) * N * DIM;
  int rowBase = blockIdx.x * 16;
  int tid = threadIdx.x, wid = tid >> 5, lane = tid & 31;

  for (int idx = tid; idx < 16 * DIM; idx += 512)
    sA[idx] = A1[(size_t)(rowBase + (idx >> 6)) * DIM + (idx & 63)];
  __syncthreads();

  v2f afrag[16];
  load_afrag(sA, lane, afrag);
  int koff = (lane < 16) ? 0 : 2;

  double accL = 0.0;
  int ntile = (N + 15) / 16;
  for (int t = wid; t < ntile; t += 16) {
    int cb = t * 16;
    int jcol = cb + (lane & 15);
    bool jvalid = jcol < N;
    const float* Brow = A2 + (size_t)(jvalid ? jcol : 0) * DIM;  // clamped, loads safe
    v8f c = tile16x16(afrag, Brow, koff);
    if (jvalid) {
      #pragma unroll
      for (int r = 0; r < 8; ++r) accL += (double)expf(c[r]);
    }
  }
  accL = waveSumD(accL);
  if (lane == 0) redD[wid] = accL;
  __syncthreads();
  if (tid == 0) {
    double s = 0.0;
    for (int w = 0; w < 16; ++w) s += redD[w];
    atomicAdd(&acc[ACC_E12 + d], s);
  }
}

// ---------------- masked cross gathers: M12, M21 ----------------
__global__ void k_gather(const float* __restrict__ norm, const int* __restrict__ topkIdx,
                         int N, double* __restrict__ acc) {
  int d = blockIdx.y;
  const float* A1 = norm + (size_t)(2 * d)     * N * DIM;
  const float* A2 = norm + (size_t)(2 * d + 1) * N * DIM;
  const int* tk1 = topkIdx + (size_t)(2 * d)     * N * KTOP;
  const int* tk2 = topkIdx + (size_t)(2 * d + 1) * N * KTOP;
  int wid = threadIdx.x >> 5, lane = threadIdx.x & 31;
  int i = blockIdx.x * 8 + wid;
  if (i >= N) return;
  size_t base = (size_t)i * DIM + lane * 2;
  float a1x = A1[base], a1y = A1[base + 1];
  float a2x = A2[base], a2y = A2[base + 1];
  double s12 = 0.0, s21 = 0.0;
  for (int t = 0; t < KTOP; ++t) {
    int j2 = tk2[(size_t)i * KTOP + t];
    float bx = A2[(size_t)j2 * DIM + lane * 2], by = A2[(size_t)j2 * DIM + lane * 2 + 1];
    float v = waveSumF(a1x * bx + a1y * by);     // S12[i, j2]
    s12 += (double)expf(v);
    int j1 = tk1[(size_t)i * KTOP + t];
    bx = A1[(size_t)j1 * DIM + lane * 2]; by = A1[(size_t)j1 * DIM + lane * 2 + 1];
    v = waveSumF(a2x * bx + a2y * by);           // S12[j1, i] = a2[i].a1[j1]
    s21 += (double)expf(v);
  }
  if (lane == 0) {
    atomicAdd(&acc[ACC_M12 + d], s12);
    atomicAdd(&acc[ACC_M21 + d], s21);
  }
}

// ---------------- self term: sum exp(x*x / 0.1) per matrix ----------------
__global__ void k_selfterm(const float* __restrict__ norm, int N, double* __restrict__ acc) {
  __shared__ double redD[8];
  int m = blockIdx.y;
  const float* X = norm + (size_t)m * N * DIM;
  int total = N * DIM;
  double a = 0.0;
  for (int idx = blockIdx.x * blockDim.x + threadIdx.x; idx < total;
       idx += gridDim.x * blockDim.x) {
    float x = X[idx];
    a += (double)expf((x * x) / 0.1f);
  }
  a = waveSumD(a);
  int wid = threadIdx.x >> 5, lane = threadIdx.x & 31;
  if (lane == 0) redD[wid] = a;
  __syncthreads();
  if (threadIdx.x == 0) {
    double s = 0.0;
    for (int w = 0; w < 8; ++w) s += redD[w];
    atomicAdd(&acc[ACC_ST + m], s);
  }
}

// ---------------- finalize ----------------
__global__ void k_final(const double* __restrict__ acc, float* __restrict__ out, int N) {
  if (threadIdx.x != 0 || blockIdx.x != 0) return;
  double Nd = (double)N;
  double Z = Nd * Nd - Nd * (double)KTOP;   // count of masked-out entries, each exp(0)=1
  double total = 0.0;
  for (int d = 0; d < 2; ++d) {
    int m0 = 2 * d, m1 = 2 * d + 1;
    double E12 = acc[ACC_E12 + d];
    double t1a = acc[ACC_E + m0] - (acc[ACC_T + m0] + Z) + acc[ACC_ST + m0];
    double t2a = E12 - (acc[ACC_M12 + d] + Z);
    total += -Nd * log(1.0 + t1a + t2a);     // t0/t0 == 1
    double t1b = acc[ACC_E + m1] - (acc[ACC_T + m1] + Z) + acc[ACC_ST + m1];
    double t2b = E12 - (acc[ACC_M21 + d] + Z);
    total += -Nd * log(1.0 + t1b + t2b);
  }
  out[0] = (float)(total * 0.25);
}

extern "C" void kernel_launch(void* const* d_in, const int* in_sizes, int n_in,
                              void* d_out, int out_size, void* d_ws, size_t ws_size,
                              hipStream_t stream) {
  const float* u1 = (const float*)d_in[0];
  const float* u2 = (const float*)d_in[1];
  const float* i1 = (const float*)d_in[2];
  const float* i2 = (const float*)d_in[3];
  int N = in_sizes[0] / DIM;   // 6000

  char* ws = (char*)d_ws;
  size_t normBytes = (size_t)4 * N * DIM * sizeof(float);
  size_t topkBytes = (size_t)4 * N * KTOP * sizeof(int);
  float* norm = (float*)ws;
  int*   topk = (int*)(ws + normBytes);
  double* acc = (double*)(ws + ((normBytes + topkBytes + 7) & ~(size_t)7));

  hipMemsetAsync(acc, 0, 18 * sizeof(double), stream);

  k_normalize<<<dim3((N + 7) / 8, 4), dim3(256), 0, stream>>>(u1, u2, i1, i2, norm, N);

  size_t smem = (size_t)16 * PSTR * 4 + (size_t)16 * DIM * 4 +
                (size_t)16 * CANDN * 4 + (size_t)16 * CANDN * 4 + 16 * 8;
  hipFuncSetAttribute(reinterpret_cast<const void*>(k_selfsim),
                      hipFuncAttributeMaxDynamicSharedMemorySize, (int)smem);
  k_selfsim<<<dim3(N / 16, 4), dim3(512), smem, stream>>>(norm, N, topk, acc);

  k_cross<<<dim3(N / 16, 2), dim3(512), 0, stream>>>(norm, N, acc);

  k_gather<<<dim3((N + 7) / 8, 2), dim3(256), 0, stream>>>(norm, topk, N, acc);

  k_selfterm<<<dim3(64, 4), dim3(256), 0, stream>>>(norm, N, acc);

  k_final<<<1, 32, 0, stream>>>(acc, (float*)d_out, N);
}